// SkipGramModel_28252294873515
// MI455X (gfx1250) — compile-verified
//
#include <hip/hip_runtime.h>

// Problem constants (from reference)
#define V_SZ   100000
#define D_DIM  128
#define B_POS  16384
#define CTX    20
#define N_NEG  81920
#define NSAMP  (B_POS + N_NEG)     // 98304
#define NTILES (NSAMP / 16)        // 6144 tiles of 16 samples

typedef float v2f __attribute__((ext_vector_type(2)));
typedef float v8f __attribute__((ext_vector_type(8)));

__device__ __forceinline__ float log_sigmoid(float x) {
    // stable: min(x,0) - log1p(exp(-|x|))
    float m = fminf(x, 0.0f);
    return m - log1pf(expf(-fabsf(x)));
}

// One workgroup = 16 waves = 16 samples (one 16x16 WMMA tile).
// Each wave gathers+pools one sample's context rows fully coalesced
// (float4 per lane = 512B per row per wave), stages emb_u/emb_v in LDS,
// then wave 0 computes the 16x16 Gram tile via 32x v_wmma_f32_16x16x4_f32
// and extracts the diagonal = per-sample dot products.
__global__ void __launch_bounds__(512)
sg_score_kernel(const float* __restrict__ u_w, const float* __restrict__ v_w,
                const int* __restrict__ pos_u, const int* __restrict__ pos_v,
                const int* __restrict__ neg_u, const int* __restrict__ neg_v,
                float* __restrict__ scores)
{
    __shared__ float ldsU[16 * D_DIM];
    __shared__ float ldsV[16 * D_DIM];

    const int lane = threadIdx.x & 31;
    const int wv   = threadIdx.x >> 5;        // wave id 0..15 = sample-in-tile
    const int tile = blockIdx.x;
    const int s    = tile * 16 + wv;          // global sample id

    // ---- gather + sum-pool (coalesced: lane*4 floats within the row) ----
    const int* ctx;
    int vidx;
    if (s < B_POS) {                          // uniform per tile (B_POS % 16 == 0)
        ctx  = pos_u + (size_t)s * CTX;
        vidx = pos_v[s];
    } else {
        int t = s - B_POS;
        ctx  = neg_u + (size_t)t * CTX;
        vidx = neg_v[t];
    }

    float4 acc = make_float4(0.f, 0.f, 0.f, 0.f);
#pragma unroll
    for (int c = 0; c < CTX; ++c) {
        int idx = ctx[c];                     // scalar (wave-uniform) load
        const float4 r = *(const float4*)(u_w + (size_t)idx * D_DIM + lane * 4);
        acc.x += r.x; acc.y += r.y; acc.z += r.z; acc.w += r.w;
    }
    *(float4*)(ldsU + wv * D_DIM + lane * 4) = acc;
    *(float4*)(ldsV + wv * D_DIM + lane * 4) =
        *(const float4*)(v_w + (size_t)vidx * D_DIM + lane * 4);
    __syncthreads();

    // ---- wave 0: 16x16 Gram tile via WMMA f32 16x16x4, K = 128 ----
    if (wv == 0) {
        v8f cacc = {};                        // 16x16 f32 accumulator (8 VGPRs)
        const int row  = lane & 15;           // A: M index / B: N index
        const int koff = (lane >> 4) * 2;     // ISA 16x4 layout: K{0,1} lanes 0-15, K{2,3} lanes 16-31
#pragma unroll
        for (int k = 0; k < 32; ++k) {        // 32 * K4 = 128
            v2f a = *(const v2f*)(ldsU + row * D_DIM + 4 * k + koff);
            v2f b = *(const v2f*)(ldsV + row * D_DIM + 4 * k + koff);
            cacc = __builtin_amdgcn_wmma_f32_16x16x4_f32(
                false, a, false, b, (short)0, cacc, false, false);
        }

        // diag extract: (i,i) lives at VGPR (i&7), lane i (i<8) or lane i+8 (i>=8)
        float val = cacc[0];
#pragma unroll
        for (int j = 1; j < 8; ++j)
            if ((lane & 7) == j) val = cacc[j];

        int i = -1;
        if (lane < 8)                  i = lane;        // M = 0..7
        else if (lane >= 16 && lane < 24) i = lane - 8; // M = 8..15
        if (i >= 0) {
            int g = tile * 16 + i;
            scores[g] = (g < B_POS) ? log_sigmoid(val) : log_sigmoid(-val);
        }
    }
}

// Deterministic final reduction: fixed-order strided partials + LDS tree.
__global__ void __launch_bounds__(1024)
sg_reduce_kernel(const float* __restrict__ scores, float* __restrict__ out, int n)
{
    __shared__ float sm[1024];
    float p = 0.f;
    for (int i = threadIdx.x; i < n; i += 1024)
        p += scores[i];
    sm[threadIdx.x] = p;
    __syncthreads();
#pragma unroll
    for (int s = 512; s > 0; s >>= 1) {
        if ((int)threadIdx.x < s) sm[threadIdx.x] += sm[threadIdx.x + s];
        __syncthreads();
    }
    if (threadIdx.x == 0) out[0] = -sm[0];
}

extern "C" void kernel_launch(void* const* d_in, const int* in_sizes, int n_in,
                              void* d_out, int out_size, void* d_ws, size_t ws_size,
                              hipStream_t stream)
{
    (void)in_sizes; (void)n_in; (void)out_size; (void)ws_size;
    const float* u_w   = (const float*)d_in[0];
    const float* v_w   = (const float*)d_in[1];
    const int*   pos_u = (const int*)d_in[2];
    const int*   pos_v = (const int*)d_in[3];
    const int*   neg_u = (const int*)d_in[4];
    const int*   neg_v = (const int*)d_in[5];

    float* scores = (float*)d_ws;             // NSAMP floats = 384 KB scratch

    sg_score_kernel<<<NTILES, 512, 0, stream>>>(u_w, v_w, pos_u, pos_v,
                                                neg_u, neg_v, scores);
    sg_reduce_kernel<<<1, 1024, 0, stream>>>(scores, (float*)d_out, NSAMP);
}